// Model_11828339933505
// MI455X (gfx1250) — compile-verified
//
#include <hip/hip_runtime.h>

#define EMB     300
#define N_NODES 100000
#define F_SEG   16384
#define D_DANG  8192
#define E_EDGE  4096
#define TN_TILES 19              // ceil(300/16)

typedef __attribute__((ext_vector_type(2))) float v2f;
typedef __attribute__((ext_vector_type(8))) float v8f;

// ---------------------------------------------------------------------------
// ws layout (floats unless noted):
//   frag_mean : F*EMB        (accumulated as sums, then divided in place)
//   counts    : F
//   h         : D*EMB
//   pm9       : 9*EMB*EMB
//   out0      : E*EMB
//   ccnt      : 16 ints (9 used)
//   clist     : 9*E ints
// ---------------------------------------------------------------------------

__global__ void k_zero(float* fs, int* ccnt, long n) {
    long i = (long)blockIdx.x * blockDim.x + threadIdx.x;
    long stride = (long)gridDim.x * blockDim.x;
    for (long j = i; j < n; j += stride) fs[j] = 0.0f;
    if (i < 9) ccnt[i] = 0;
}

__global__ void k_count(const int* __restrict__ fb, float* cnt) {
    long i = (long)blockIdx.x * blockDim.x + threadIdx.x;
    long stride = (long)gridDim.x * blockDim.x;
    for (long n = i; n < N_NODES; n += stride)
        atomicAdd(&cnt[fb[n]], 1.0f);
}

__global__ void k_segsum(const float* __restrict__ out, const int* __restrict__ fb,
                         float* fs) {
    long total = (long)N_NODES * EMB;
    long i = (long)blockIdx.x * blockDim.x + threadIdx.x;
    long stride = (long)gridDim.x * blockDim.x;
    for (long j = i; j < total; j += stride) {
        int n = (int)(j / EMB);
        int c = (int)(j - (long)n * EMB);
        atomicAdd(&fs[(long)fb[n] * EMB + c], out[j]);
    }
}

__global__ void k_mean(float* fs, const float* __restrict__ cnt) {
    long total = (long)F_SEG * EMB;
    long i = (long)blockIdx.x * blockDim.x + threadIdx.x;
    long stride = (long)gridDim.x * blockDim.x;
    for (long j = i; j < total; j += stride)
        fs[j] /= fmaxf(cnt[j / EMB], 1.0f);
}

__global__ void k_pm9(const float* __restrict__ e1, const float* __restrict__ e2,
                      float* __restrict__ pm9) {
    const long M = (long)EMB * EMB;
    long total = 9 * M;
    long i = (long)blockIdx.x * blockDim.x + threadIdx.x;
    long stride = (long)gridDim.x * blockDim.x;
    for (long j = i; j < total; j += stride) {
        int c = (int)(j / M);
        long r = j - (long)c * M;
        pm9[j] = e1[(long)(c / 3) * M + r] + e2[(long)(c % 3) * M + r];
    }
}

__global__ void k_bucket(const int* __restrict__ attr, int* ccnt, int* clist) {
    int e = blockIdx.x * blockDim.x + threadIdx.x;
    if (e >= E_EDGE) return;
    int a0 = attr[2 * e + 0], a1 = attr[2 * e + 1];
    int c = a0 * 3 + a1;                 // both in [0,3) -> 9 classes
    int pos = atomicAdd(&ccnt[c], 1);
    clist[c * E_EDGE + pos] = e;
}

// h = out[dangling_idx] @ W_proj + b_proj + mean[fb[dangling_idx]]
// Boundary handling: output (m,n) depends only on A row m / B col n, so
// out-of-range columns are handled by CLAMPING the B pointer (garbage flows
// only into lanes that skip the store) -- no guarded loads, no exec churn.
__global__ void k_gemm_h(const float* __restrict__ out, const int* __restrict__ didx,
                         const int* __restrict__ fb, const float* __restrict__ W,
                         const float* __restrict__ bproj, const float* __restrict__ mean,
                         float* __restrict__ h) {
    int wid  = threadIdx.x >> 5;
    int lane = threadIdx.x & 31;
    int tile = blockIdx.x * (blockDim.x >> 5) + wid;
    const int total = (D_DANG / 16) * TN_TILES;
    if (tile >= total) return;
    int tm = tile / TN_TILES, tn = tile % TN_TILES;
    int m0 = tm * 16, n0 = tn * 16;
    int half = lane >> 4;                // selects K pair {0,1} vs {2,3}
    int l15  = lane & 15;

    int node = didx[m0 + l15];
    const float* aptr = out + (long)node * EMB + half * 2;

    int  ncol = n0 + l15;
    bool nok  = ncol < EMB;
    int  ncc  = nok ? ncol : (EMB - 1);              // clamped column
    const float* bptr = W + (long)(half * 2) * EMB + ncc;

    v8f acc = {};
#pragma unroll 5
    for (int k0 = 0; k0 < EMB; k0 += 4) {
        v2f a, b;
        a.x = aptr[k0];
        a.y = aptr[k0 + 1];
        b.x = bptr[(long)k0 * EMB];
        b.y = bptr[(long)(k0 + 1) * EMB];
        __builtin_prefetch(bptr + (long)(k0 + 20) * EMB, 0, 1);
        acc = __builtin_amdgcn_wmma_f32_16x16x4_f32(
            false, a, false, b, (short)0, acc, false, false);
    }

    if (nok) {
        float bp = bproj[ncol];
#pragma unroll
        for (int j = 0; j < 8; ++j) {
            int m   = half ? (j + 8) : j;
            int row = m0 + m;
            int f   = fb[didx[row]];
            h[(long)row * EMB + ncol] = acc[j] + bp + mean[(long)f * EMB + ncol];
        }
    }
}

// out0[e] = h[ei0[e]] @ pm9[class(e)]  -- 9 GEMMs over class-bucketed edges.
// Rows beyond cnt are clamped to the last valid class row (garbage only
// reaches output rows that skip the store).
__global__ void k_edge_gemm(const float* __restrict__ h, const int* __restrict__ ei0,
                            const int* __restrict__ clist, const int* __restrict__ ccnt,
                            const float* __restrict__ pm9, float* __restrict__ out0) {
    const int TM = E_EDGE / 16;          // 256 worst-case row tiles per class
    int wid  = threadIdx.x >> 5;
    int lane = threadIdx.x & 31;
    int tile = blockIdx.x * (blockDim.x >> 5) + wid;
    const int total = 9 * TM * TN_TILES;
    if (tile >= total) return;
    int c   = tile / (TM * TN_TILES);
    int rem = tile % (TM * TN_TILES);
    int tm  = rem / TN_TILES, tn = rem % TN_TILES;
    int cnt = ccnt[c];
    int m0  = tm * 16;
    if (m0 >= cnt) return;               // cnt >= 1 past this point
    int n0 = tn * 16;

    int half = lane >> 4;
    int l15  = lane & 15;

    int ridx = m0 + l15;
    if (ridx >= cnt) ridx = cnt - 1;     // clamp row (no guarded loads)
    int e    = clist[c * E_EDGE + ridx];
    int node = ei0[e];
    const float* aptr = h + (long)node * EMB + half * 2;

    int  ncol = n0 + l15;
    bool nok  = ncol < EMB;
    int  ncc  = nok ? ncol : (EMB - 1);
    const float* bptr = pm9 + (long)c * EMB * EMB + (long)(half * 2) * EMB + ncc;

    v8f acc = {};
#pragma unroll 5
    for (int k0 = 0; k0 < EMB; k0 += 4) {
        v2f a, b;
        a.x = aptr[k0];
        a.y = aptr[k0 + 1];
        b.x = bptr[(long)k0 * EMB];
        b.y = bptr[(long)(k0 + 1) * EMB];
        __builtin_prefetch(bptr + (long)(k0 + 20) * EMB, 0, 1);
        acc = __builtin_amdgcn_wmma_f32_16x16x4_f32(
            false, a, false, b, (short)0, acc, false, false);
    }

    if (nok) {
#pragma unroll
        for (int j = 0; j < 8; ++j) {
            int m = half ? (j + 8) : j;
            int r = m0 + m;
            if (r < cnt) {
                int ee = clist[c * E_EDGE + r];
                out0[(long)ee * EMB + ncol] = acc[j];
            }
        }
    }
}

// logits[e]   = dot(out0[e], h[ei1[e]])
// logits[E+e] = dot(out0[e], h[ei1[(e-1) mod E]])   (roll by SHIFT_K=1)
__global__ void k_logits(const float* __restrict__ out0, const float* __restrict__ h,
                         const int* __restrict__ ei1, float* __restrict__ outp) {
    int wid  = threadIdx.x >> 5;
    int lane = threadIdx.x & 31;
    int e = blockIdx.x * (blockDim.x >> 5) + wid;
    if (e >= E_EDGE) return;
    const float* o  = out0 + (long)e * EMB;
    const float* h1 = h + (long)ei1[e] * EMB;
    int ep = (e + E_EDGE - 1) % E_EDGE;
    const float* h2 = h + (long)ei1[ep] * EMB;
    float s1 = 0.0f, s2 = 0.0f;
    for (int i = lane; i < EMB; i += 32) {
        float ov = o[i];
        s1 += ov * h1[i];
        s2 += ov * h2[i];
    }
#pragma unroll
    for (int off = 16; off > 0; off >>= 1) {
        s1 += __shfl_xor(s1, off);
        s2 += __shfl_xor(s2, off);
    }
    if (lane == 0) {
        outp[e]          = s1;
        outp[E_EDGE + e] = s2;
    }
}

__global__ void k_labels(float* outp) {
    int i = blockIdx.x * blockDim.x + threadIdx.x;
    if (i < 2 * E_EDGE) outp[2 * E_EDGE + i] = (i < E_EDGE) ? 1.0f : 0.0f;
}

extern "C" void kernel_launch(void* const* d_in, const int* in_sizes, int n_in,
                              void* d_out, int out_size, void* d_ws, size_t ws_size,
                              hipStream_t stream) {
    const float* out_nodes = (const float*)d_in[0];   // [N, EMB]
    const int*   fb        = (const int*)d_in[1];     // [N]
    const int*   didx      = (const int*)d_in[2];     // [D]
    const int*   eidx      = (const int*)d_in[3];     // [2, E]
    const int*   attr      = (const int*)d_in[4];     // [E, 2]
    const float* W         = (const float*)d_in[5];   // [EMB, EMB]
    const float* bproj     = (const float*)d_in[6];   // [EMB]
    const float* e1        = (const float*)d_in[7];   // [6, EMB*EMB]
    const float* e2        = (const float*)d_in[8];   // [3, EMB*EMB]
    float*       outp      = (float*)d_out;           // logits[2E] ++ labels[2E]

    const int* ei0 = eidx;
    const int* ei1 = eidx + E_EDGE;

    // workspace carve-up
    float* fs    = (float*)d_ws;                       // F*EMB (sum -> mean)
    float* cntf  = fs + (size_t)F_SEG * EMB;           // F
    float* h     = cntf + F_SEG;                       // D*EMB
    float* pm9   = h + (size_t)D_DANG * EMB;           // 9*EMB*EMB
    float* out0  = pm9 + (size_t)9 * EMB * EMB;        // E*EMB
    int*   ccnt  = (int*)(out0 + (size_t)E_EDGE * EMB);// 16 ints (9 used)
    int*   clist = ccnt + 16;                          // 9*E ints

    // 1. zero accumulators (frag sums + counts contiguous) and class counters
    long nz = (long)F_SEG * EMB + F_SEG;
    k_zero<<<(int)((nz + 255) / 256), 256, 0, stream>>>(fs, ccnt, nz);

    // 2. segment counts + sums
    k_count<<<512, 256, 0, stream>>>(fb, cntf);
    k_segsum<<<4096, 256, 0, stream>>>(out_nodes, fb, fs);

    // 3. mean in place
    k_mean<<<4096, 256, 0, stream>>>(fs, cntf);

    // 4. build 9 combined edge matrices
    k_pm9<<<2048, 256, 0, stream>>>(e1, e2, pm9);

    // 5. bucket edges by (attr0, attr1) class
    k_bucket<<<(E_EDGE + 255) / 256, 256, 0, stream>>>(attr, ccnt, clist);

    // 6. h = out[didx] @ W + b + mean[fb[didx]]   (WMMA f32)
    {
        int tiles = (D_DANG / 16) * TN_TILES;          // 512 * 19
        int blocks = (tiles + 3) / 4;                  // 4 waves / block
        k_gemm_h<<<blocks, 128, 0, stream>>>(out_nodes, didx, fb, W, bproj, fs, h);
    }

    // 7. out0 = h[ei0] @ pm9[class]                  (WMMA f32, 9 GEMMs)
    {
        int tiles = 9 * (E_EDGE / 16) * TN_TILES;      // 9 * 256 * 19
        int blocks = (tiles + 3) / 4;
        k_edge_gemm<<<blocks, 128, 0, stream>>>(h, ei0, clist, ccnt, pm9, out0);
    }

    // 8. logits (one wave per edge) + labels
    k_logits<<<(E_EDGE + 7) / 8, 256, 0, stream>>>(out0, h, ei1, outp);
    k_labels<<<(2 * E_EDGE + 255) / 256, 256, 0, stream>>>(outp);
}